// ReverseLSTMLayer_45226005627002
// MI455X (gfx1250) — compile-verified
//
#include <hip/hip_runtime.h>
#include <math.h>

// MI455X / gfx1250, wave32. fp32 LSTM.
// Phase 1 (x_proj GEMM, 68.7 GFLOP, 320 MB traffic): ~15us HBM-bound -> LDS-tiled
//   V_WMMA_F32_16X16X4_F32 with async global->LDS staging (ASYNCcnt path).
// Phase 2 (512 sequential steps): latency bound -> persistent kernel, W_hh slice
//   resident in LDS all 512 steps, one grid barrier per step, batched L2 loads of h.

typedef __attribute__((ext_vector_type(2))) float v2f;
typedef __attribute__((ext_vector_type(8))) float v8f;

#define T_DIM 512
#define B_DIM 64
#define I_DIM 512
#define H_DIM 512
#define G_DIM 2048  // 4*H

// workspace byte offsets
#define XPROJ_OFF 0ull                 // 512*64*2048 f32 = 256 MB
#define WTIH_OFF  268435456ull         // [I][4H] f32 = 4 MB
#define WTHH_OFF  272629760ull         // [H][4H] f32 = 4 MB
#define BIAS_OFF  276824064ull         // [4H] f32
#define HBUF_OFF  276832256ull         // 2*[B][H] f32 (ping-pong h state)
#define BAR_OFF   277094400ull         // grid barrier counter

// ---- CDNA5 async global->LDS (ASYNCcnt) with safe fallback --------------------
#if defined(__has_builtin)
#if __has_builtin(__builtin_amdgcn_global_load_async_to_lds_b128) && \
    __has_builtin(__builtin_amdgcn_s_wait_asynccnt)
#define USE_ASYNC_LDS 1
#endif
#endif

typedef int v4i_vs __attribute__((__vector_size__(16)));  // matches builtin's pointee type

__device__ __forceinline__ void copy16_g2l(const float* g, float* l) {
#ifdef USE_ASYNC_LDS
  __builtin_amdgcn_global_load_async_to_lds_b128(
      (__attribute__((address_space(1))) v4i_vs*)g,
      (__attribute__((address_space(3))) v4i_vs*)l, 0, 0);
#else
  *(float4*)l = *(const float4*)g;
#endif
}
__device__ __forceinline__ void copy16_wait() {
#ifdef USE_ASYNC_LDS
  __builtin_amdgcn_s_wait_asynccnt(0);
#endif
}
// -------------------------------------------------------------------------------

__device__ __forceinline__ float sigmoidf_(float x) { return 1.0f / (1.0f + __expf(-x)); }

// ---------------------------------------------------------------------------
// Prep: transpose weights to K-major, fuse biases, seed h ping-pong slot 0.
// ---------------------------------------------------------------------------
__global__ void prep_kernel(const float* __restrict__ Wih, const float* __restrict__ Whh,
                            const float* __restrict__ bih, const float* __restrict__ bhh,
                            const float* __restrict__ h0,
                            float* __restrict__ wTih, float* __restrict__ wThh,
                            float* __restrict__ bias, float* __restrict__ hbuf) {
  const int WN = G_DIM * I_DIM;  // 1M
  int id = blockIdx.x * blockDim.x + threadIdx.x;
  if (id < WN) {
    int k = id / G_DIM, g = id % G_DIM;
    wTih[id] = Wih[g * I_DIM + k];
  } else if (id < 2 * WN) {
    int id2 = id - WN;
    int k = id2 / G_DIM, g = id2 % G_DIM;
    wThh[id2] = Whh[g * H_DIM + k];
  } else if (id < 2 * WN + G_DIM) {
    int g = id - 2 * WN;
    bias[g] = bih[g] + bhh[g];
  } else if (id < 2 * WN + G_DIM + B_DIM * H_DIM) {
    int i = id - 2 * WN - G_DIM;
    hbuf[i] = h0[i];
  }
}

// ---------------------------------------------------------------------------
// Phase 1: x_proj[m][g] = input[m][:] . wTih[:][g] + bias[g]
// 64x64 block tile, 8 waves x 2 N-tiles. K staged in 32-wide chunks via 16B
// async copies. A stride 36 (16B aligned, 16 distinct banks); B stride 68.
// Fragments preloaded into register arrays so DS waits batch instead of
// serializing against each WMMA.
// ---------------------------------------------------------------------------
__global__ void __launch_bounds__(256) xproj_kernel(const float* __restrict__ x,
                                                    const float* __restrict__ wT,
                                                    const float* __restrict__ bias,
                                                    float* __restrict__ xp) {
  __shared__ float lds_a[64 * 36];  // [m][k], KC=32, pad 4 (b128-aligned)
  __shared__ float lds_b[32 * 68];  // [k][n], NB=64, pad 4
  const int M0 = blockIdx.x * 64, N0 = blockIdx.y * 64;
  const int tid = threadIdx.x;
  const int wave = tid >> 5, lane = tid & 31;
  const int mt = wave & 3, ntb = (wave >> 2) * 2;
  const int half = lane >> 4, ln = lane & 15;
  v8f acc0 = {}; v8f acc1 = {};

  for (int kc = 0; kc < I_DIM; kc += 32) {
    // stage A: 64x32 f32 = 512 float4, 2 per thread (8 threads/row -> 128B bursts)
#pragma unroll
    for (int r = 0; r < 2; r++) {
      int idx = tid + r * 256;
      int mrow = idx >> 3, k4 = (idx & 7) * 4;
      copy16_g2l(&x[(size_t)(M0 + mrow) * I_DIM + kc + k4], &lds_a[mrow * 36 + k4]);
    }
    // stage B: 32x64 f32 = 512 float4, 2 per thread (16 threads/row -> 256B bursts)
#pragma unroll
    for (int r = 0; r < 2; r++) {
      int idx = tid + r * 256;
      int krow = idx >> 4, n4 = (idx & 15) * 4;
      copy16_g2l(&wT[(size_t)(kc + krow) * G_DIM + N0 + n4], &lds_b[krow * 68 + n4]);
    }
    copy16_wait();
    __syncthreads();

    // preload fragments for the whole chunk, then run the WMMA chain
    v2f af[8], bf0[8], bf1[8];
#pragma unroll
    for (int i = 0; i < 8; i++) {
      int kb = i * 4 + 2 * half;
      af[i] = *(const v2f*)&lds_a[(mt * 16 + ln) * 36 + kb];
    }
#pragma unroll
    for (int i = 0; i < 8; i++) {
      int kb = i * 4 + 2 * half;
      bf0[i][0] = lds_b[kb * 68 + ntb * 16 + ln];
      bf0[i][1] = lds_b[(kb + 1) * 68 + ntb * 16 + ln];
      bf1[i][0] = lds_b[kb * 68 + (ntb + 1) * 16 + ln];
      bf1[i][1] = lds_b[(kb + 1) * 68 + (ntb + 1) * 16 + ln];
    }
#pragma unroll
    for (int i = 0; i < 8; i++) {
      acc0 = __builtin_amdgcn_wmma_f32_16x16x4_f32(false, af[i], false, bf0[i], (short)0, acc0, false, false);
      acc1 = __builtin_amdgcn_wmma_f32_16x16x4_f32(false, af[i], false, bf1[i], (short)0, acc1, false, false);
    }
    __syncthreads();
  }
  // C/D layout: elem e -> row M = e + 8*half, col N = ln
#pragma unroll
  for (int e = 0; e < 8; e++) {
    int m = M0 + mt * 16 + e + 8 * half;
    int n0 = N0 + ntb * 16 + ln;
    int n1 = N0 + (ntb + 1) * 16 + ln;
    xp[(size_t)m * G_DIM + n0] = acc0[e] + bias[n0];
    xp[(size_t)m * G_DIM + n1] = acc1[e] + bias[n1];
  }
}

// ---------------------------------------------------------------------------
// Phase 2: persistent reverse scan. 64 blocks; block wg owns H-cols [wg*8,wg*8+8)
// (gate columns {q*512 + wg*8 + j}). W_hh slice (512x32, stride 33 -> conflict
// free) resident in LDS for all 512 steps; c-slice in LDS; h ping-pongs through
// an L2-resident global buffer with ONE grid barrier per step. Per-step GEMM
// runs 64-wide K chunks with 16 global b64 A-loads batched in flight.
// ---------------------------------------------------------------------------
__global__ void __launch_bounds__(256) lstm_kernel(const float* __restrict__ xp,
                                                   const float* __restrict__ wT,
                                                   const float* __restrict__ c0,
                                                   float* __restrict__ hbuf,
                                                   float* __restrict__ out,
                                                   unsigned* __restrict__ bar) {
  __shared__ float lds_w[512 * 33];
  __shared__ float lds_g[64 * 32];
  __shared__ float lds_c[64 * 8];
  const int wg = blockIdx.x;  // 0..63
  const int hc0 = wg * 8;
  const int tid = threadIdx.x;
  const int wave = tid >> 5, lane = tid & 31;
  const int mt = wave & 3, nt = wave >> 2;  // 4 m-tiles x 2 n-tiles
  const int half = lane >> 4, ln = lane & 15;
  const unsigned nblocks = gridDim.x;

  // stage W_hh slice: local col lc -> gate q=lc/8, j=lc%8 -> g = q*512 + hc0 + j
  for (int idx = tid; idx < 512 * 32; idx += 256) {
    int k = idx >> 5, lc = idx & 31;
    int g = (lc >> 3) * H_DIM + hc0 + (lc & 7);
    lds_w[k * 33 + lc] = wT[(size_t)k * G_DIM + g];
  }
  for (int idx = tid; idx < 64 * 8; idx += 256) {
    int b = idx >> 3, j = idx & 7;
    lds_c[idx] = c0[b * H_DIM + hc0 + j];
  }
  __syncthreads();

  const int m = mt * 16 + ln;   // batch row for this lane's A fragment
  const int lc = nt * 16 + ln;  // local gate column 0..31
  const int g = (lc >> 3) * H_DIM + hc0 + (lc & 7);

  for (int s = 0; s < T_DIM; ++s) {
    const int t = T_DIM - 1 - s;
    const float* hprev = hbuf + (size_t)(s & 1) * (B_DIM * H_DIM);
    float* hnext = hbuf + (size_t)((s & 1) ^ 1) * (B_DIM * H_DIM);

    v8f acc0 = {}; v8f acc1 = {};  // two accumulators break the WMMA RAW chain
    for (int k0 = 0; k0 < H_DIM; k0 += 64) {
      v2f af[16], bf[16];
#pragma unroll
      for (int i = 0; i < 16; i++) {  // 16 L2 loads issued back-to-back
        int kb = k0 + i * 4 + 2 * half;
        af[i] = *(const v2f*)&hprev[(size_t)m * H_DIM + kb];
      }
#pragma unroll
      for (int i = 0; i < 16; i++) {
        int kb = k0 + i * 4 + 2 * half;
        bf[i][0] = lds_w[kb * 33 + lc];
        bf[i][1] = lds_w[(kb + 1) * 33 + lc];
      }
#pragma unroll
      for (int i = 0; i < 16; i += 2) {
        acc0 = __builtin_amdgcn_wmma_f32_16x16x4_f32(false, af[i],     false, bf[i],     (short)0, acc0, false, false);
        acc1 = __builtin_amdgcn_wmma_f32_16x16x4_f32(false, af[i + 1], false, bf[i + 1], (short)0, acc1, false, false);
      }
    }

    // add precomputed input projection; park gates in LDS for the exchange
    const float* xrow = xp + (size_t)t * B_DIM * G_DIM;
#pragma unroll
    for (int e = 0; e < 8; e++) {
      int mr = mt * 16 + e + 8 * half;
      lds_g[mr * 32 + lc] = acc0[e] + acc1[e] + xrow[(size_t)mr * G_DIM + g];
    }
    __syncthreads();

    // elementwise LSTM cell: 64 rows x 8 cols = 512 items, 2 per thread
#pragma unroll
    for (int r = 0; r < 2; r++) {
      int item = tid + r * 256;
      int b = item >> 3, j = item & 7;
      float ig = sigmoidf_(lds_g[b * 32 + j]);
      float fg = sigmoidf_(lds_g[b * 32 + 8 + j]);
      float gg = tanhf(lds_g[b * 32 + 16 + j]);
      float og = sigmoidf_(lds_g[b * 32 + 24 + j]);
      float cn = fg * lds_c[item] + ig * gg;
      float hn = og * tanhf(cn);
      lds_c[item] = cn;
      hnext[b * H_DIM + hc0 + j] = hn;
      out[(size_t)t * B_DIM * H_DIM + b * H_DIM + hc0 + j] = hn;
      if (s == T_DIM - 1) {  // final state (after t==0)
        out[(size_t)T_DIM * B_DIM * H_DIM + b * H_DIM + hc0 + j] = hn;                  // h_f
        out[(size_t)T_DIM * B_DIM * H_DIM + B_DIM * H_DIM + b * H_DIM + hc0 + j] = cn;  // c_f
      }
    }

    // grid-wide barrier: monotonic counter, zeroed by hipMemsetAsync each launch
    __threadfence();
    __syncthreads();
    if (tid == 0) {
      atomicAdd(bar, 1u);
      unsigned target = (unsigned)(s + 1) * nblocks;
      while (__hip_atomic_load(bar, __ATOMIC_RELAXED, __HIP_MEMORY_SCOPE_AGENT) < target) {
        __builtin_amdgcn_s_sleep(1);
      }
      __threadfence();
    }
    __syncthreads();
  }
}

extern "C" void kernel_launch(void* const* d_in, const int* in_sizes, int n_in,
                              void* d_out, int out_size, void* d_ws, size_t ws_size,
                              hipStream_t stream) {
  const float* input = (const float*)d_in[0];
  const float* h0    = (const float*)d_in[1];
  const float* c0    = (const float*)d_in[2];
  const float* Wih   = (const float*)d_in[3];
  const float* Whh   = (const float*)d_in[4];
  const float* bih   = (const float*)d_in[5];
  const float* bhh   = (const float*)d_in[6];
  float* out = (float*)d_out;
  char* ws = (char*)d_ws;
  float* xpj  = (float*)(ws + XPROJ_OFF);
  float* wTih = (float*)(ws + WTIH_OFF);
  float* wThh = (float*)(ws + WTHH_OFF);
  float* bias = (float*)(ws + BIAS_OFF);
  float* hbuf = (float*)(ws + HBUF_OFF);
  unsigned* bar = (unsigned*)(ws + BAR_OFF);

  (void)hipMemsetAsync(bar, 0, 128, stream);

  int prepN = 2 * G_DIM * I_DIM + G_DIM + B_DIM * H_DIM;
  prep_kernel<<<(prepN + 255) / 256, 256, 0, stream>>>(Wih, Whh, bih, bhh, h0,
                                                       wTih, wThh, bias, hbuf);

  dim3 g1(T_DIM * B_DIM / 64, G_DIM / 64);  // 512 x 32 blocks
  xproj_kernel<<<g1, 256, 0, stream>>>(input, wTih, bias, xpj);

  lstm_kernel<<<64, 256, 0, stream>>>(xpj, wThh, c0, hbuf, out, bar);
}